// ModelNew_7868380086953
// MI455X (gfx1250) — compile-verified
//
#include <hip/hip_runtime.h>
#include <hip/hip_bf16.h>
#include <stdint.h>

// Problem constants (from reference): B=4, T=16, H=32, D=128, L=4096, HALF=64
#define BT          64          // B*T
#define HD          4096        // H*D floats per (b, pos) row
#define HD4         1024        // HD in float4 units
#define L_CACHE     4096
#define NK_FLOATS   67108864ull // B*L*H*D
#define NK_F4       16777216ull // NK_FLOATS / 4
#define TOTAL_F4    33554432ull // 2 * NK_F4
#define F4_PER_BLK  2048        // 256 threads * 8 float4
#define COPY_BLOCKS 16384       // TOTAL_F4 / F4_PER_BLK (exact, no tail)

typedef float v4f __attribute__((ext_vector_type(4)));

// ---------------------------------------------------------------------------
// Kernel 1: bulk cache -> output copy via the CDNA5 async LDS copy engine.
// GVS addressing: wave-uniform 64-bit base in SGPR pair, 32-bit per-lane
// VGPR offsets (no 64-bit VALU address math). Streaming NT cache policy on
// both sides (536 MB each way vs 192 MB L2: zero reuse).
// Each block stages exactly 32 KB through LDS:
//   8 async b128 loads/thread -> s_wait_asynccnt 0 -> 8 async b128 stores.
// ---------------------------------------------------------------------------
__global__ __launch_bounds__(256) void bulk_copy_async(
    const float* __restrict__ srcK,
    const float* __restrict__ srcV,
    float* __restrict__ dst)
{
    __shared__ __align__(16) char lbuf[F4_PER_BLK * 16];  // 32 KB staging

    const uint32_t tid  = threadIdx.x;
    const uint64_t blk4 = (uint64_t)blockIdx.x * (uint64_t)F4_PER_BLK;

    // Block is entirely within the K half or entirely within the V half
    // (NK_F4 is a multiple of F4_PER_BLK), so this select is wave-uniform.
    uint64_t sbase;
    if (blk4 < NK_F4) sbase = (uint64_t)(uintptr_t)srcK + blk4 * 16ull;
    else              sbase = (uint64_t)(uintptr_t)srcV + (blk4 - NK_F4) * 16ull;
    const uint64_t dbase = (uint64_t)(uintptr_t)dst + blk4 * 16ull;

    // Low 32 bits of a generic pointer to LDS == wave-relative LDS offset.
    const uint32_t lds0 = (uint32_t)(uintptr_t)(&lbuf[0]);
    const uint32_t t16  = tid * 16u;

    // Phase 1: global -> LDS (async, ASYNCcnt-tracked, saddr form, NT)
#pragma unroll
    for (int k = 0; k < 8; ++k) {
        const uint32_t voff = t16 + (uint32_t)(k * 4096);  // global byte offset
        const uint32_t l    = lds0 + voff;                 // LDS byte address
        asm volatile("global_load_async_to_lds_b128 %0, %1, %2 th:TH_LOAD_NT"
                     :: "v"(l), "v"(voff), "s"(sbase) : "memory");
    }
    asm volatile("s_wait_asynccnt 0" ::: "memory");

    // Phase 2: LDS -> global (async, saddr form, NT)
#pragma unroll
    for (int k = 0; k < 8; ++k) {
        const uint32_t voff = t16 + (uint32_t)(k * 4096);
        const uint32_t l    = lds0 + voff;
        asm volatile("global_store_async_from_lds_b128 %0, %1, %2 th:TH_STORE_NT"
                     :: "v"(voff), "v"(l), "s"(dbase) : "memory");
    }
    // Drain before LDS/block retirement (S_ENDPGM also implies wait-idle).
    asm volatile("s_wait_asynccnt 0" ::: "memory");
}

// ---------------------------------------------------------------------------
// Kernel 2: RoPE-rotate k_new and scatter k_rot / v_new into the output
// caches at per-(b,t) positions. One block per (b,t). Interleaved RoPE pairs
// are adjacent in memory, so one float4 load carries two (even,odd) pairs.
// Runs after the bulk copy on the same stream, so its writes win (matching
// .at[].set() semantics; positions are unique).
// ---------------------------------------------------------------------------
__global__ __launch_bounds__(256) void rope_scatter(
    const float* __restrict__ k_new,
    const float* __restrict__ v_new,
    const float* __restrict__ cosp,
    const float* __restrict__ sinp,
    const int*   __restrict__ positions,
    float* __restrict__ outK,
    float* __restrict__ outV)
{
    const int bt  = blockIdx.x;      // 0..63
    const int b   = bt >> 4;         // T = 16
    const int pos = positions[bt];   // unique, in [0, L)

    __shared__ float cs[64];
    __shared__ float sn[64];
    if (threadIdx.x < 64) {
        cs[threadIdx.x] = cosp[(size_t)pos * 64 + threadIdx.x];
        sn[threadIdx.x] = sinp[(size_t)pos * 64 + threadIdx.x];
    }
    __syncthreads();

    const v4f* __restrict__ kin = (const v4f*)(k_new) + (size_t)bt * HD4;
    const v4f* __restrict__ vin = (const v4f*)(v_new) + (size_t)bt * HD4;
    const size_t row4 = ((size_t)b * L_CACHE + (size_t)pos) * HD4;
    v4f* __restrict__ ko = (v4f*)(outK) + row4;
    v4f* __restrict__ vo = (v4f*)(outV) + row4;

    // f indexes float4 within the (b,t) row: f = h*32 + j, j in [0,32)
#pragma unroll
    for (int it = 0; it < 4; ++it) {
        const int f = it * 256 + threadIdx.x;   // 0..1023
        const int j = f & 31;                   // float4 index within head
        const v4f kv = kin[f];
        const float c0 = cs[2 * j],     s0 = sn[2 * j];
        const float c1 = cs[2 * j + 1], s1 = sn[2 * j + 1];
        v4f o;
        o.x = kv.x * c0 - kv.y * s0;   // even' = e*c - o*s
        o.y = kv.x * s0 + kv.y * c0;   // odd'  = e*s + o*c
        o.z = kv.z * c1 - kv.w * s1;
        o.w = kv.z * s1 + kv.w * c1;
        ko[f] = o;
        vo[f] = vin[f];
    }
}

// ---------------------------------------------------------------------------
extern "C" void kernel_launch(void* const* d_in, const int* in_sizes, int n_in,
                              void* d_out, int out_size, void* d_ws, size_t ws_size,
                              hipStream_t stream) {
    (void)in_sizes; (void)n_in; (void)out_size; (void)d_ws; (void)ws_size;

    const float* k_new     = (const float*)d_in[0];
    const float* v_new     = (const float*)d_in[1];
    const float* cosp      = (const float*)d_in[2];
    const float* sinp      = (const float*)d_in[3];
    const float* cache_k   = (const float*)d_in[4];
    const float* cache_v   = (const float*)d_in[5];
    const int*   positions = (const int*)d_in[6];

    float* out  = (float*)d_out;
    float* outK = out;                 // [B, L, H, D]
    float* outV = out + NK_FLOATS;     // [B, L, H, D]

    // 1) bulk copy caches -> output (streaming, async LDS path, NT policy)
    bulk_copy_async<<<COPY_BLOCKS, 256, 0, stream>>>(cache_k, cache_v, out);
    // 2) RoPE + scatter overwrite of the 64 touched rows (stream-ordered)
    rope_scatter<<<BT, 256, 0, stream>>>(k_new, v_new, cosp, sinp, positions,
                                         outK, outV);
}